// BoundleAdjustment_2302102470656
// MI455X (gfx1250) — compile-verified
//
#include <hip/hip_runtime.h>
#include <math.h>

#define BLOCK    256
#define NBLOCKS  512
#define NCAM     32

typedef __attribute__((ext_vector_type(2))) float v2f;
typedef __attribute__((ext_vector_type(8))) float v8f;

// Wave-wide sum of one float per lane, computed on the matrix unit.
// A(16x4) holds the 32 lane accumulators (K=0 for lanes 0-15, K=2 for lanes
// 16-31, other K slots zero); B(4x16) = ones. D[i][j] = acc_i + acc_{i+16}
// for every column j, so each lane sums its 8 D VGPRs (rows 0-7 / 8-15 of the
// row-sums) and one shfl_xor(16) yields the full wave total in every lane.
// EXEC must be all ones at the call site (WMMA requirement).
__device__ __forceinline__ float wave_sum32(float acc) {
    v2f a; a[0] = acc;  a[1] = 0.0f;
    v2f b; b[0] = 1.0f; b[1] = 1.0f;
    v8f c = {};
    c = __builtin_amdgcn_wmma_f32_16x16x4_f32(
        /*neg_a=*/false, a, /*neg_b=*/false, b,
        /*c_mod=*/(short)0, c, /*reuse_a=*/false, /*reuse_b=*/false);
    float s = c[0] + c[1] + c[2] + c[3] + c[4] + c[5] + c[6] + c[7];
    s += __shfl_xor(s, 16, 32);
    return s;
}

__global__ __launch_bounds__(BLOCK)
void ba_main_kernel(const float* __restrict__ pole,      // [2]
                    const float* __restrict__ p3d,       // [M,3,3]
                    const float* __restrict__ p2d,       // [M,C,3,2]
                    const int*   __restrict__ mask,      // [M,C]
                    const float* __restrict__ Km,        // [C,3,3]
                    const float* __restrict__ distm,     // [C,5]
                    const float* __restrict__ Rm,        // [C,3,3]
                    const float* __restrict__ tm,        // [C,3]
                    float*       __restrict__ partials,  // [NBLOCKS]
                    int M)
{
    const int lane = threadIdx.x & 31;

    // ---- per-lane camera constants (lane == camera index) ----
    const float* Rc = Rm + lane * 9;
    const float r00 = Rc[0], r01 = Rc[1], r02 = Rc[2];
    const float r10 = Rc[3], r11 = Rc[4], r12 = Rc[5];
    const float r20 = Rc[6], r21 = Rc[7], r22 = Rc[8];
    const float t0 = tm[lane * 3 + 0], t1 = tm[lane * 3 + 1], t2 = tm[lane * 3 + 2];
    const float fx = Km[lane * 9 + 0], u0 = Km[lane * 9 + 2];
    const float fy = Km[lane * 9 + 4], v0 = Km[lane * 9 + 5];
    const float k1  = distm[lane * 5 + 0];
    const float k2  = distm[lane * 5 + 1];
    const float pp1 = distm[lane * 5 + 2];
    const float pp2 = distm[lane * 5 + 3];
    const float k3  = distm[lane * 5 + 4];

    const float s  = pole[0] + pole[1];
    const float wa = pole[1] / s;     // weight on point 0
    const float wb = pole[0] / s;     // weight on point 2

    // uniform (SGPR) wave index for grid-stride over poles
    const int wave  = __builtin_amdgcn_readfirstlane(
                          (blockIdx.x * BLOCK + threadIdx.x) >> 5);
    const int nwave = gridDim.x * (BLOCK >> 5);

    float lacc = 0.0f;

    auto proj_err = [&](float px, float py, float pz, float qx, float qy) -> float {
        float X0 = fmaf(r00, px, fmaf(r01, py, fmaf(r02, pz, t0)));
        float X1 = fmaf(r10, px, fmaf(r11, py, fmaf(r12, pz, t1)));
        float X2 = fmaf(r20, px, fmaf(r21, py, fmaf(r22, pz, t2)));
        float inv = 1.0f / X2;
        float x0 = X0 * inv, x1 = X1 * inv;
        float r2 = x0 * x0 + x1 * x1;
        float radial = 1.0f + k1 * r2 + k2 * r2 * r2 + k3 * r2 * r2 * r2;
        float xd = x0 * radial + 2.0f * pp1 * x0 * x1 + pp2 * (r2 + 2.0f * x0 * x0);
        float yd = x1 * radial + pp1 * (r2 + 2.0f * x1 * x1) + 2.0f * pp2 * x0 * x1;
        float du = qx - fmaf(fx, xd, u0);
        float dv = qy - fmaf(fy, yd, v0);
        return sqrtf(du * du + dv * dv);
    };

    for (int m = wave; m < M; m += nwave) {
        // pole_3ds[m] : wave-uniform 9-float read (scalarizable)
        const float* __restrict__ P = p3d + (size_t)m * 9;
        const float p0x = P[0], p0y = P[1], p0z = P[2];
        const float p1x = P[3], p1y = P[4], p1z = P[5];
        const float p2x = P[6], p2y = P[7], p2z = P[8];

        // this lane's observations: 6 contiguous floats (8B-aligned),
        // single-use -> non-temporal streaming loads
        const float* q = p2d + ((size_t)m * NCAM + lane) * 6;
        v2f q01 = __builtin_nontemporal_load((const v2f*)(q + 0));
        v2f q23 = __builtin_nontemporal_load((const v2f*)(q + 2));
        v2f q45 = __builtin_nontemporal_load((const v2f*)(q + 4));
        int msk = __builtin_nontemporal_load(&mask[(size_t)m * NCAM + lane]);

        float sum3 = proj_err(p0x, p0y, p0z, q01[0], q01[1])
                   + proj_err(p1x, p1y, p1z, q23[0], q23[1])
                   + proj_err(p2x, p2y, p2z, q45[0], q45[1]);

        lacc += msk ? 0.01f * sum3 : 0.0f;

        if (lane == 0) {
            // loss_line = || p1 - (wa*p0 + wb*p2) ||
            float ex = p1x - (wa * p0x + wb * p2x);
            float ey = p1y - (wa * p0y + wb * p2y);
            float ez = p1z - (wa * p0z + wb * p2z);
            float ll = sqrtf(ex * ex + ey * ey + ez * ez);
            // loss_len = | ||p0 - p2|| - s |
            float dx = p0x - p2x, dy = p0y - p2y, dz = p0z - p2z;
            float ln = fabsf(sqrtf(dx * dx + dy * dy + dz * dz) - s);
            lacc += 0.01f * (ll + ln);
        }

        // lookahead for the next grid-stride iteration (global_prefetch_b8)
        if (m + nwave < M) {
            __builtin_prefetch(p2d + ((size_t)(m + nwave) * NCAM + lane) * 6, 0, 1);
        }
    }

    // ---- wave reduction on the matrix unit, then block combine ----
    float wtot = wave_sum32(lacc);

    __shared__ float wsum[BLOCK / 32];
    const int wid = threadIdx.x >> 5;
    if (lane == 0) wsum[wid] = wtot;
    __syncthreads();
    if (wid == 0) {
        float v = (lane < (BLOCK / 32)) ? wsum[lane] : 0.0f;
        v += __shfl_xor(v, 4, 32);
        v += __shfl_xor(v, 2, 32);
        v += __shfl_xor(v, 1, 32);
        if (lane == 0) partials[blockIdx.x] = v;
    }
}

__global__ __launch_bounds__(256)
void ba_reduce_kernel(const float* __restrict__ partials,
                      float* __restrict__ out, int n, float invM)
{
    const int lane = threadIdx.x & 31;
    const int wid  = threadIdx.x >> 5;

    float acc = 0.0f;
    for (int i = threadIdx.x; i < n; i += 256) acc += partials[i];

    float wtot = wave_sum32(acc);

    __shared__ float wsum[8];
    if (lane == 0) wsum[wid] = wtot;
    __syncthreads();
    if (wid == 0) {
        float v = (lane < 8) ? wsum[lane] : 0.0f;
        v += __shfl_xor(v, 4, 32);
        v += __shfl_xor(v, 2, 32);
        v += __shfl_xor(v, 1, 32);
        if (lane == 0) out[0] = v * invM;
    }
}

extern "C" void kernel_launch(void* const* d_in, const int* in_sizes, int n_in,
                              void* d_out, int out_size, void* d_ws, size_t ws_size,
                              hipStream_t stream) {
    const float* pole  = (const float*)d_in[0];   // [2]
    const float* p3d   = (const float*)d_in[1];   // [M,3,3]
    const float* p2d   = (const float*)d_in[2];   // [M,C,3,2]
    const int*   mask  = (const int*)  d_in[3];   // [M,C]
    const float* Km    = (const float*)d_in[4];   // [C,3,3]
    const float* distm = (const float*)d_in[5];   // [C,5]
    const float* Rm    = (const float*)d_in[6];   // [C,3,3]
    const float* tm    = (const float*)d_in[7];   // [C,3]

    const int M = in_sizes[1] / 9;
    float* partials = (float*)d_ws;               // NBLOCKS floats

    ba_main_kernel<<<NBLOCKS, BLOCK, 0, stream>>>(
        pole, p3d, p2d, mask, Km, distm, Rm, tm, partials, M);
    ba_reduce_kernel<<<1, 256, 0, stream>>>(
        partials, (float*)d_out, NBLOCKS, 1.0f / (float)M);
}